// InferenceModel_69097433858784
// MI455X (gfx1250) — compile-verified
//
#include <hip/hip_runtime.h>
#include <hip/hip_bf16.h>

// ---------------------------------------------------------------------------
// RetinaNet-style post-process for MI455X (gfx1250, wave32).
// Pipeline: decode -> per-(b,c) top-1000 -> greedy NMS -> per-image top-100.
// CDNA5 paths used: global_load_async_to_lds_b128 (+ s_wait_asynccnt) to DMA
// the NMS box tile into LDS, global_prefetch_b8 in the streaming scans, and
// LDS histogram/bitonic machinery (ds_* ops).
// ---------------------------------------------------------------------------

#define AS1 __attribute__((address_space(1)))
#define AS3 __attribute__((address_space(3)))

#ifndef __has_builtin
#define __has_builtin(x) 0
#endif

typedef int v4i __attribute__((ext_vector_type(4)));

static constexpr int   BB    = 4;
static constexpr int   NN    = 49104;
static constexpr int   CC    = 80;
static constexpr int   KP    = 1000;   // max_box_pre_nms
static constexpr int   KPAD  = 1024;   // padded for bitonic sort
static constexpr int   MAXB  = 100;
static constexpr float THR   = 0.01f;
static constexpr float IOU_T = 0.5f;
static constexpr float WIMG  = 512.0f;
static constexpr int   NBKT  = 4096;   // score histogram buckets over (0,1]

// ---- CDNA5 async copy helpers (guarded; fallback keeps compile clean) -----
__device__ __forceinline__ void async_copy_b128(const void* gsrc, void* ldst) {
#if __has_builtin(__builtin_amdgcn_global_load_async_to_lds_b128)
    // builtin signature: (global int4* src, local int4* dst, imm off, imm cpol)
    v4i* g = (v4i*)gsrc;   // drop const, generic pointee cast
    v4i* l = (v4i*)ldst;
    __builtin_amdgcn_global_load_async_to_lds_b128(
        (AS1 v4i*)g, (AS3 v4i*)l, 0, 0);
#else
    *(float4*)ldst = *(const float4*)gsrc;
#endif
}

__device__ __forceinline__ void wait_async0() {
#if __has_builtin(__builtin_amdgcn_s_wait_asynccnt)
    __builtin_amdgcn_s_wait_asynccnt(0);
#else
    asm volatile("s_wait_asynccnt 0" ::: "memory");
#endif
}

__device__ __forceinline__ float sigmoidf_(float x) {
    return 1.0f / (1.0f + __expf(-x));
}

// ---------------------------------------------------------------------------
// Kernel 1: BBoxTransform + ClipBoxes.  anchors yxyx -> boxes xyxy clipped.
// ---------------------------------------------------------------------------
__global__ __launch_bounds__(256) void decode_kernel(
    const float* __restrict__ anchors, const float* __restrict__ regression,
    float* __restrict__ boxes, int total) {
    int i = blockIdx.x * blockDim.x + threadIdx.x;
    if (i >= total) return;
    float4 a = ((const float4*)anchors)[i];     // y1 x1 y2 x2
    float4 r = ((const float4*)regression)[i];  // dy dx dh dw
    float yca = (a.x + a.z) * 0.5f;
    float xca = (a.y + a.w) * 0.5f;
    float ha  = a.z - a.x;
    float wa  = a.w - a.y;
    float w   = __expf(r.w) * wa;
    float h   = __expf(r.z) * ha;
    float yc  = r.x * ha + yca;
    float xc  = r.y * wa + xca;
    float4 o;
    o.x = fmaxf(xc - w * 0.5f, 0.0f);
    o.y = fmaxf(yc - h * 0.5f, 0.0f);
    o.z = fminf(xc + w * 0.5f, WIMG - 1.0f);
    o.w = fminf(yc + h * 0.5f, WIMG - 1.0f);
    ((float4*)boxes)[i] = o;
}

// ---------------------------------------------------------------------------
// Kernel 2: per-(image,class) threshold + top-1000 (sorted desc).
// One workgroup (8 wave32) per (b,c).  Histogram-select then bitonic sort.
// ---------------------------------------------------------------------------
__global__ __launch_bounds__(256) void topk_kernel(
    const float* __restrict__ cls, const float* __restrict__ boxes,
    float* __restrict__ vals, float* __restrict__ bsel) {
    __shared__ int   hist[NBKT];
    __shared__ float ssc[KPAD];
    __shared__ int   sidx[KPAD];
    __shared__ int   scount;
    __shared__ int   cutoff;

    const int bc = blockIdx.x;
    const int b = bc / CC, c = bc % CC;
    const int tid = threadIdx.x;
    const float* colbase = cls + (size_t)b * NN * CC + c;

    for (int i = tid; i < NBKT; i += 256) hist[i] = 0;
    __syncthreads();

    // Pass 1: histogram of sigmoid scores above threshold.
    for (int n = tid; n < NN; n += 256) {
        __builtin_prefetch(colbase + (size_t)(n + 2048) * CC, 0, 0); // global_prefetch_b8
        float p = sigmoidf_(colbase[(size_t)n * CC]);
        if (p > THR) {
            int bkt = min((int)(p * (float)NBKT), NBKT - 1);
            atomicAdd(&hist[bkt], 1);
        }
    }
    __syncthreads();

    if (tid == 0) {
        int cum = 0, cut = 0;
        for (int bkt = NBKT - 1; bkt >= 0; --bkt) {
            cum += hist[bkt];
            if (cum >= KP) { cut = bkt; break; }
        }
        cutoff = cut;
        scount = 0;
    }
    for (int i = tid; i < KPAD; i += 256) { ssc[i] = -1.0f; sidx[i] = -1; }
    __syncthreads();

    // Pass 2: collect candidates at/above the cutoff bucket (L2-resident).
    const int cut = cutoff;
    for (int n = tid; n < NN; n += 256) {
        float p = sigmoidf_(colbase[(size_t)n * CC]);
        if (p > THR) {
            int bkt = min((int)(p * (float)NBKT), NBKT - 1);
            if (bkt >= cut) {
                int pos = atomicAdd(&scount, 1);
                if (pos < KPAD) { ssc[pos] = p; sidx[pos] = n; }
            }
        }
    }
    __syncthreads();

    // Bitonic sort, 1024 wide: score desc, index asc tiebreak (matches top_k).
    for (int k = 2; k <= KPAD; k <<= 1) {
        for (int j = k >> 1; j > 0; j >>= 1) {
            for (int t = tid; t < KPAD; t += 256) {
                int ixj = t ^ j;
                if (ixj > t) {
                    float s0 = ssc[t], s1 = ssc[ixj];
                    int   i0 = sidx[t], i1 = sidx[ixj];
                    bool ixj_first = (s1 > s0) || (s1 == s0 && i1 < i0);
                    bool t_first   = (s0 > s1) || (s0 == s1 && i0 < i1);
                    bool descBlock = ((t & k) == 0);
                    if (descBlock ? ixj_first : t_first) {
                        ssc[t] = s1; ssc[ixj] = s0;
                        sidx[t] = i1; sidx[ixj] = i0;
                    }
                }
            }
            __syncthreads();
        }
    }

    // Emit sorted scores + gathered boxes.
    const size_t row = (size_t)bc * KP;
    for (int t = tid; t < KP; t += 256) {
        float s = ssc[t];
        int   id = sidx[t];
        vals[row + t] = s;
        float4 bx = make_float4(0.f, 0.f, 0.f, 0.f);
        if (id >= 0) bx = ((const float4*)boxes)[(size_t)b * NN + id];
        ((float4*)bsel)[row + t] = bx;
    }
}

// ---------------------------------------------------------------------------
// Kernel 3: greedy NMS per (b,c) row.  Boxes staged in LDS via CDNA5 async
// global->LDS DMA (ASYNCcnt tracked), 16 KB per row.
// ---------------------------------------------------------------------------
__global__ __launch_bounds__(256) void nms_kernel(
    const float* __restrict__ vals, const float* __restrict__ bsel,
    int* __restrict__ keep) {
    __shared__ float4 sbox[KP];   // 16 KB
    __shared__ int    skeep[KP];  //  4 KB
    const int bc = blockIdx.x, tid = threadIdx.x;
    const size_t row = (size_t)bc * KP;
    const float4* gb = ((const float4*)bsel) + row;

    for (int t = tid; t < KP; t += 256) {
        async_copy_b128(&gb[t], &sbox[t]);           // global_load_async_to_lds_b128
        skeep[t] = (vals[row + t] > THR) ? 1 : 0;
    }
    wait_async0();                                    // s_wait_asynccnt 0
    __syncthreads();

    for (int i = 0; i < KP; ++i) {
        __syncthreads();
        if (!skeep[i]) continue;                      // uniform branch
        const float4 bi = sbox[i];
        const float area_i = (bi.z - bi.x) * (bi.w - bi.y);
        for (int j = i + 1 + tid; j < KP; j += 256) {
            if (!skeep[j]) continue;
            float4 bj = sbox[j];
            float x1 = fmaxf(bi.x, bj.x), y1 = fmaxf(bi.y, bj.y);
            float x2 = fminf(bi.z, bj.z), y2 = fminf(bi.w, bj.w);
            float inter = fmaxf(x2 - x1, 0.0f) * fmaxf(y2 - y1, 0.0f);
            float area_j = (bj.z - bj.x) * (bj.w - bj.y);
            float iou = inter / (area_i + area_j - inter);
            if (iou > IOU_T) skeep[j] = 0;
        }
    }
    __syncthreads();
    for (int t = tid; t < KP; t += 256) keep[row + t] = skeep[t];
}

// ---------------------------------------------------------------------------
// Kernel 4: per-image merge of 80*1000 candidates -> sorted top-100 + output.
// Output layout (floats): frois[B,100,4] | fcls[B,100] | fsc[B,100].
// ---------------------------------------------------------------------------
__global__ __launch_bounds__(256) void final_kernel(
    const float* __restrict__ vals, const int* __restrict__ keep,
    const float* __restrict__ bsel, const float* __restrict__ scale,
    float* __restrict__ out) {
    __shared__ int   hist[NBKT];
    __shared__ float ssc[256];
    __shared__ int   sidx[256];
    __shared__ int   scount;
    __shared__ int   cutoff;

    const int b = blockIdx.x, tid = threadIdx.x;
    const size_t base = (size_t)b * CC * KP;
    const int TOT = CC * KP;

    for (int i = tid; i < NBKT; i += 256) hist[i] = 0;
    __syncthreads();

    for (int f = tid; f < TOT; f += 256) {
        float v = vals[base + f];
        if (v > THR && keep[base + f]) {
            int bkt = min((int)(v * (float)NBKT), NBKT - 1);
            atomicAdd(&hist[bkt], 1);
        }
    }
    __syncthreads();

    if (tid == 0) {
        int cum = 0, cut = 0;
        for (int bkt = NBKT - 1; bkt >= 0; --bkt) {
            cum += hist[bkt];
            if (cum >= MAXB) { cut = bkt; break; }
        }
        cutoff = cut;
        scount = 0;
    }
    if (tid < 256) { ssc[tid] = -1.0f; sidx[tid] = -1; }
    __syncthreads();

    const int cut = cutoff;
    for (int f = tid; f < TOT; f += 256) {
        float v = vals[base + f];
        if (v > THR && keep[base + f]) {
            int bkt = min((int)(v * (float)NBKT), NBKT - 1);
            if (bkt >= cut) {
                int pos = atomicAdd(&scount, 1);
                if (pos < 256) { ssc[pos] = v; sidx[pos] = f; }
            }
        }
    }
    __syncthreads();

    // Bitonic sort 256 (one element per thread): score desc, flat idx asc.
    for (int k = 2; k <= 256; k <<= 1) {
        for (int j = k >> 1; j > 0; j >>= 1) {
            int t = tid, ixj = t ^ j;
            if (ixj > t) {
                float s0 = ssc[t], s1 = ssc[ixj];
                int   i0 = sidx[t], i1 = sidx[ixj];
                bool ixj_first = (s1 > s0) || (s1 == s0 && i1 < i0);
                bool t_first   = (s0 > s1) || (s0 == s1 && i0 < i1);
                bool descBlock = ((t & k) == 0);
                if (descBlock ? ixj_first : t_first) {
                    ssc[t] = s1; ssc[ixj] = s0;
                    sidx[t] = i1; sidx[ixj] = i0;
                }
            }
            __syncthreads();
        }
    }

    const float sc = scale[b];
    if (tid < MAXB) {
        float v = ssc[tid];
        int   f = sidx[tid];
        bool valid = (v > THR) && (f >= 0);
        float4 bx = make_float4(0.f, 0.f, 0.f, 0.f);
        int cls = -1;
        if (valid) {
            float4 raw = ((const float4*)bsel)[base + f];
            bx = make_float4(raw.x / sc, raw.y / sc, raw.z / sc, raw.w / sc);
            cls = f / KP;
        }
        const size_t ro = (size_t)b * MAXB + tid;
        ((float4*)out)[ro] = bx;                                       // frois
        out[(size_t)BB * MAXB * 4 + ro] = (float)cls;                  // fcls
        out[(size_t)BB * MAXB * 4 + (size_t)BB * MAXB + ro] =
            valid ? v : 0.0f;                                          // fsc
    }
}

// ---------------------------------------------------------------------------
extern "C" void kernel_launch(void* const* d_in, const int* in_sizes, int n_in,
                              void* d_out, int out_size, void* d_ws, size_t ws_size,
                              hipStream_t stream) {
    (void)in_sizes; (void)n_in; (void)out_size; (void)ws_size;
    // setup_inputs order: image(0, unused), anchors(1), regression(2),
    //                     classification(3), scale(4)
    const float* anchors        = (const float*)d_in[1];
    const float* regression     = (const float*)d_in[2];
    const float* classification = (const float*)d_in[3];
    const float* scale          = (const float*)d_in[4];

    char* ws = (char*)d_ws;
    size_t off = 0;
    float* boxes = (float*)(ws + off); off += (size_t)BB * NN * 4 * sizeof(float);      // 3.14 MB
    float* vals  = (float*)(ws + off); off += (size_t)BB * CC * KP * sizeof(float);     // 1.28 MB
    float* bsel  = (float*)(ws + off); off += (size_t)BB * CC * KP * 4 * sizeof(float); // 5.12 MB
    int*   keep  = (int*)(ws + off);   off += (size_t)BB * CC * KP * sizeof(int);       // 1.28 MB

    const int total = BB * NN;
    decode_kernel<<<(total + 255) / 256, 256, 0, stream>>>(anchors, regression, boxes, total);
    topk_kernel  <<<BB * CC, 256, 0, stream>>>(classification, boxes, vals, bsel);
    nms_kernel   <<<BB * CC, 256, 0, stream>>>(vals, bsel, keep);
    final_kernel <<<BB, 256, 0, stream>>>(vals, keep, bsel, scale, (float*)d_out);
}